// Ggnn_25391846653986
// MI455X (gfx1250) — compile-verified
//
#include <hip/hip_runtime.h>

// GGNN message kernel for MI455X (gfx1250, wave32):
//   out[r,:] = edge_matrix[label[r]] @ h_w[r,:]   (800k rows, 64x64, 8 labels)
//
// Strategy: per-block label binning + fp32 WMMA (V_WMMA_F32_16X16X4_F32).
// Groups are built label-major; each wave takes a contiguous slice of groups
// so the 64 v2f B fragments (one 64x64 edge matrix in WMMA layout, 128 VGPRs)
// are loaded once per label change and reused across ~9 groups from registers.
// h_w / out are pure single-use streams -> non-temporal (TH=NT) so they don't
// evict the L2-resident 128 KB edge-matrix table.

typedef __attribute__((ext_vector_type(2))) float v2f;
typedef __attribute__((ext_vector_type(8))) float v8f;

#define CHUNK      1024
#define PADSLOP    128                 // 8 labels * up to 15 pad entries
#define MAXGROUPS  (CHUNK / 16 + 8)
#define NWAVES     8
#define INVALID    0xFFFFFFFFu

__global__ __launch_bounds__(256)
void ggnn_msg_wmma(const float* __restrict__ hw,     // [rows, 64]
                   const int*   __restrict__ labels, // [rows]
                   const float* __restrict__ emat,   // [8, 64(out), 64(in)]
                   float*       __restrict__ out,    // [rows, 64]
                   int rows)
{
    __shared__ unsigned sCnt[8];
    __shared__ unsigned sFill[8];
    __shared__ unsigned sBase[8];
    __shared__ unsigned sIdx[CHUNK + PADSLOP];
    __shared__ unsigned sGroup[MAXGROUPS];   // (label<<16) | bin offset
    __shared__ unsigned sNumGroups;

    const int tid       = threadIdx.x;
    const int chunkBase = blockIdx.x * CHUNK;
    const int n         = min(CHUNK, rows - chunkBase);

    if (tid < 8) { sCnt[tid] = 0u; sFill[tid] = 0u; }
    __syncthreads();

    // Pass 1: histogram labels; fill index bins with sentinel.
    for (int i = tid; i < CHUNK + PADSLOP; i += 256) sIdx[i] = INVALID;
    for (int i = tid; i < n; i += 256) {
        unsigned lab = (unsigned)labels[chunkBase + i];
        if (lab < 8u) atomicAdd(&sCnt[lab], 1u);
    }
    __syncthreads();

    // Serial prefix over 8 labels; emit one descriptor per 16-row group,
    // label-major so consecutive groups usually share a label.
    if (tid == 0) {
        unsigned base = 0u, ng = 0u;
        for (int l = 0; l < 8; ++l) {
            sBase[l] = base;
            unsigned c      = sCnt[l];
            unsigned padded = (c + 15u) & ~15u;
            for (unsigned off = 0u; off < padded; off += 16u)
                sGroup[ng++] = ((unsigned)l << 16) | (base + off);
            base += padded;
        }
        sNumGroups = ng;
    }
    __syncthreads();

    // Pass 2: scatter row indices into their label's bin.
    for (int i = tid; i < n; i += 256) {
        unsigned lab = (unsigned)labels[chunkBase + i];
        if (lab < 8u) {
            unsigned pos = atomicAdd(&sFill[lab], 1u);
            sIdx[sBase[lab] + pos] = (unsigned)(chunkBase + i);
        }
    }
    __syncthreads();

    const int wave    = tid >> 5;
    const int lane    = tid & 31;
    const int halfsel = lane >> 4;    // 0: lanes 0-15, 1: lanes 16-31
    const int l15     = lane & 15;
    const int koff    = halfsel * 2;  // A/B fragment K sub-offset per ISA layout

    // Contiguous group slice per wave (wave-uniform bounds -> EXEC stays full).
    const unsigned ng   = sNumGroups;
    const unsigned gPer = (ng + NWAVES - 1u) / NWAVES;
    const unsigned gBeg = (unsigned)wave * gPer;
    const unsigned gEnd = min(ng, gBeg + gPer);

    unsigned curLab = INVALID;
    v2f bf[64];                        // B fragments for current label: [t*16+kk]

    for (unsigned g = gBeg; g < gEnd; ++g) {
        const unsigned desc = sGroup[g];
        const unsigned lab  = desc >> 16;
        const unsigned base = desc & 0xFFFFu;

        // ---- (re)load B fragments only on label change (wave-uniform) ----
        // B[k, nCol] = E[lab, o=nCol, i=k]; per-lane v0/v1 carry K pairs
        // {0,1} (lanes 0-15) / {2,3} (lanes 16-31) of each K-step.
        if (lab != curLab) {
            curLab = lab;
            const float* bp =
                emat + (size_t)lab * 4096u + (size_t)l15 * 64u + koff;
#pragma unroll
            for (int t = 0; t < 4; ++t)
#pragma unroll
                for (int kk = 0; kk < 16; ++kk)
                    bf[t * 16 + kk] = *(const v2f*)(bp + t * 1024 + kk * 4);
        }

        // ---- A fragment: 16 rows x K=64, held as 16 K-steps of 4 ----
        // Streamed once -> non-temporal so it doesn't pollute L2.
        const unsigned rowA = sIdx[base + l15];
        const float* hrow =
            hw + (size_t)(rowA == INVALID ? 0u : rowA) * 64u + koff;
        v2f a[16];
#pragma unroll
        for (int kk = 0; kk < 16; ++kk)
            a[kk] = __builtin_nontemporal_load((const v2f*)(hrow + kk * 4));

        v8f acc0 = (v8f)0.0f, acc1 = (v8f)0.0f,
            acc2 = (v8f)0.0f, acc3 = (v8f)0.0f;

#pragma unroll
        for (int kk = 0; kk < 16; ++kk) {
            acc0 = __builtin_amdgcn_wmma_f32_16x16x4_f32(
                false, a[kk], false, bf[0 * 16 + kk], (short)0, acc0, false, false);
            acc1 = __builtin_amdgcn_wmma_f32_16x16x4_f32(
                false, a[kk], false, bf[1 * 16 + kk], (short)0, acc1, false, false);
            acc2 = __builtin_amdgcn_wmma_f32_16x16x4_f32(
                false, a[kk], false, bf[2 * 16 + kk], (short)0, acc2, false, false);
            acc3 = __builtin_amdgcn_wmma_f32_16x16x4_f32(
                false, a[kk], false, bf[3 * 16 + kk], (short)0, acc3, false, false);
        }

        // ---- store: C VGPR r holds (M=r, N=lane) / (M=8+r, N=lane-16) ----
        // 16 lanes of a half-wave write 16 consecutive floats of one row.
        // Output is write-once -> non-temporal stores.
#pragma unroll
        for (int r = 0; r < 8; ++r) {
            const unsigned ro = sIdx[base + (unsigned)(r + halfsel * 8)];
            if (ro != INVALID) {
                float* orow = out + (size_t)ro * 64u + (unsigned)l15;
                __builtin_nontemporal_store(acc0[r], orow + 0);
                __builtin_nontemporal_store(acc1[r], orow + 16);
                __builtin_nontemporal_store(acc2[r], orow + 32);
                __builtin_nontemporal_store(acc3[r], orow + 48);
            }
        }
    }
}

extern "C" void kernel_launch(void* const* d_in, const int* in_sizes, int n_in,
                              void* d_out, int out_size, void* d_ws, size_t ws_size,
                              hipStream_t stream) {
    // setup_inputs order: h_v (unused), h_w, e_vw, edge_matrix
    const float* hw   = (const float*)d_in[1];
    const int*   lab  = (const int*)d_in[2];
    const float* emat = (const float*)d_in[3];
    float*       out  = (float*)d_out;

    const int rows   = in_sizes[2];                 // B*N (one label per row)
    const int blocks = (rows + CHUNK - 1) / CHUNK;

    hipLaunchKernelGGL(ggnn_msg_wmma, dim3(blocks), dim3(256), 0, stream,
                       hw, lab, emat, out, rows);
}